// SimpleLSTMModel_34437047779724
// MI455X (gfx1250) — compile-verified
//
#include <hip/hip_runtime.h>
#include <stdint.h>

// Problem constants (from reference): B=16384, S=336, F=8, H=1
#define S_LEN   336
#define F_DIM   8
#define CH      16                 // timesteps per chunk
#define NCHUNK  (S_LEN / CH)       // 21, exact
#define ROWS    32                 // batch rows per wave (one per lane)
#define RSTR    17                 // padded LDS row stride (floats) -> conflict-free
#define TILE    (ROWS * RSTR)      // 544 floats
#define WAVE_LDS (3 * TILE)        // xbuf0, xbuf1, obuf
#define NWAVE   8                  // 256-thread block = 8 wave32

// ---- fast transcendentals (v_exp_f32 is exp2; v_rcp_f32) ----
__device__ __forceinline__ float fast_sig(float x) {
    // 1 / (1 + 2^(-x*log2(e)))
    return __builtin_amdgcn_rcpf(1.0f + __builtin_amdgcn_exp2f(-1.4426950408889634f * x));
}
__device__ __forceinline__ float fast_tanh(float x) {
    // 2*sigmoid(2x) - 1
    return 2.0f * __builtin_amdgcn_rcpf(1.0f + __builtin_amdgcn_exp2f(-2.8853900817779268f * x)) - 1.0f;
}

// Low 32 bits of a generic pointer into LDS == workgroup-relative LDS byte offset
__device__ __forceinline__ unsigned lds_byte_off(const void* p) {
    return (unsigned)(uintptr_t)p;
}

// ---- CDNA5 async global<->LDS (ASYNCcnt-tracked), GVS addressing ----
__device__ __forceinline__ void async_load_b32(unsigned lds_b, unsigned goff, uint64_t sbase) {
    asm volatile("global_load_async_to_lds_b32 %0, %1, %2"
                 :: "v"(lds_b), "v"(goff), "s"(sbase) : "memory");
}
__device__ __forceinline__ void async_store_b32(unsigned goff, unsigned lds_b, uint64_t sbase) {
    asm volatile("global_store_async_from_lds_b32 %0, %1, %2"
                 :: "v"(goff), "v"(lds_b), "s"(sbase) : "memory");
}
__device__ __forceinline__ void wait_async0() { asm volatile("s_wait_asynccnt 0" ::: "memory"); }
__device__ __forceinline__ void wait_ds0()    { asm volatile("s_wait_dscnt 0"    ::: "memory"); }

__global__ __launch_bounds__(256, 1) void lstm2_fused_kernel(
    const float* __restrict__ xin,
    const float* __restrict__ Wx1, const float* __restrict__ Wh1, const float* __restrict__ B1,
    const float* __restrict__ Wx2, const float* __restrict__ Wh2, const float* __restrict__ B2,
    const float* __restrict__ fcW, const float* __restrict__ fcB,
    float* __restrict__ out)
{
    __shared__ float lds[NWAVE * WAVE_LDS];   // 8 * 1632 * 4 = 52224 B

    const unsigned tid  = threadIdx.x;
    const unsigned lane = tid & 31u;
    const unsigned wave = tid >> 5u;
    const unsigned bw0  = blockIdx.x * 256u + wave * 32u;   // first batch row of this wave

    float* wls   = &lds[wave * WAVE_LDS];
    float* xb0   = wls;
    float* xb1   = wls + TILE;
    float* obuf  = wls + 2 * TILE;

    // uniform weights (H=1 => 4 scalars per matrix)
    const float wx10=Wx1[0], wx11=Wx1[1], wx12=Wx1[2], wx13=Wx1[3];
    const float wh10=Wh1[0], wh11=Wh1[1], wh12=Wh1[2], wh13=Wh1[3];
    const float b10 =B1[0],  b11 =B1[1],  b12 =B1[2],  b13 =B1[3];
    const float wx20=Wx2[0], wx21=Wx2[1], wx22=Wx2[2], wx23=Wx2[3];
    const float wh20=Wh2[0], wh21=Wh2[1], wh22=Wh2[2], wh23=Wh2[3];
    const float b20 =B2[0],  b21 =B2[1],  b22 =B2[2],  b23 =B2[3];
    const float fw  =fcW[0], fb  =fcB[0];

    const uint64_t inBase  = (uint64_t)(uintptr_t)xin;
    const uint64_t outBase = (uint64_t)(uintptr_t)out;

    // per-lane pieces for the transfer loops: 32 lanes cover 2 rows x 16 timesteps
    const unsigned sub = lane >> 4;      // 0/1 -> which of the two rows
    const unsigned tl  = lane & 15u;     // timestep within chunk

    // issue async loads for one chunk into a buffer (16 instrs, 16x4B per row at 32B stride)
    auto issue_loads = [&](float* buf, unsigned t0) {
        #pragma unroll
        for (unsigned k = 0; k < 16; ++k) {
            const unsigned row  = 2u * k + sub;
            const unsigned goff = ((bw0 + row) * (unsigned)S_LEN + t0 + tl) * (F_DIM * 4u)
                                  + (F_DIM - 1u) * 4u;          // last feature
            async_load_b32(lds_byte_off(&buf[row * RSTR + tl]), goff, inBase);
        }
    };
    // issue async stores of obuf for one chunk (fully coalesced over t)
    auto issue_stores = [&](unsigned t0) {
        #pragma unroll
        for (unsigned k = 0; k < 16; ++k) {
            const unsigned row  = 2u * k + sub;
            const unsigned goff = ((bw0 + row) * (unsigned)S_LEN + t0 + tl) * 4u;
            async_store_b32(goff, lds_byte_off(&obuf[row * RSTR + tl]), outBase);
        }
    };

    float h1 = 0.0f, c1 = 0.0f, h2 = 0.0f, c2 = 0.0f;

    issue_loads(xb0, 0);                          // prologue: chunk 0 in flight

    for (unsigned c = 0; c < NCHUNK; ++c) {
        wait_async0();                            // loads(c) done; stores(c-1) done (obuf free)
        float* xb = (c & 1u) ? xb1 : xb0;
        if (c + 1 < NCHUNK)
            issue_loads((c & 1u) ? xb0 : xb1, (c + 1) * CH);   // overlap next chunk's fill

        const float* xrow = &xb[lane * RSTR];     // conflict-free: 17*lane mod 64 distinct
        float*       orow = &obuf[lane * RSTR];

        #pragma unroll
        for (unsigned t = 0; t < CH; ++t) {
            const float x = xrow[t];
            // --- layer 1 (H=1) ---
            float zi = fmaf(h1, wh10, fmaf(x, wx10, b10));
            float zf = fmaf(h1, wh11, fmaf(x, wx11, b11));
            float zg = fmaf(h1, wh12, fmaf(x, wx12, b12));
            float zo = fmaf(h1, wh13, fmaf(x, wx13, b13));
            c1 = fast_sig(zf) * c1 + fast_sig(zi) * fast_tanh(zg);
            h1 = fast_sig(zo) * fast_tanh(c1);
            // --- layer 2 (input = h1) ---
            zi = fmaf(h2, wh20, fmaf(h1, wx20, b20));
            zf = fmaf(h2, wh21, fmaf(h1, wx21, b21));
            zg = fmaf(h2, wh22, fmaf(h1, wx22, b22));
            zo = fmaf(h2, wh23, fmaf(h1, wx23, b23));
            c2 = fast_sig(zf) * c2 + fast_sig(zi) * fast_tanh(zg);
            h2 = fast_sig(zo) * fast_tanh(c2);
            // --- FC ---
            orow[t] = fmaf(h2, fw, fb);
        }

        wait_ds0();                               // obuf writes visible to async engine
        issue_stores(c * CH);                     // coalesced async stores, overlap next iter
    }
    wait_async0();                                // drain before endpgm
}

extern "C" void kernel_launch(void* const* d_in, const int* in_sizes, int n_in,
                              void* d_out, int out_size, void* d_ws, size_t ws_size,
                              hipStream_t stream) {
    (void)n_in; (void)out_size; (void)d_ws; (void)ws_size;
    const float* xin = (const float*)d_in[0];
    const float* Wx1 = (const float*)d_in[1];
    const float* Wh1 = (const float*)d_in[2];
    const float* B1  = (const float*)d_in[3];
    const float* Wx2 = (const float*)d_in[4];
    const float* Wh2 = (const float*)d_in[5];
    const float* B2  = (const float*)d_in[6];
    const float* fcW = (const float*)d_in[7];
    const float* fcB = (const float*)d_in[8];
    float* out = (float*)d_out;

    const int Btot = in_sizes[0] / (S_LEN * F_DIM);   // 16384
    dim3 grid((unsigned)(Btot / 256)), block(256);
    hipLaunchKernelGGL(lstm2_fused_kernel, grid, block, 0, stream,
                       xin, Wx1, Wh1, B1, Wx2, Wh2, B2, fcW, fcB, out);
}